// CTCLoss_16338055594164
// MI455X (gfx1250) — compile-verified
//
#include <hip/hip_runtime.h>

// CTC forward on MI455X (gfx1250).
// T=512, N=32, C=8000, S=40, Se=2S+1=81 states.
// One wave32 per batch element; lane l owns states 3l,3l+1,3l+2.
// Gather of log_probs[t,n,et[s]] is done with CDNA5 async global->LDS DMA
// (ASYNCcnt-tracked), 8 time-steps ahead in a 9-slot LDS ring, hiding all
// memory latency behind the serial 511-step logsumexp chain. Neighbor alpha
// exchange uses ds_bpermute (no barriers in the hot loop).

#define T_STEPS   512
#define N_BATCH   32
#define C_CLASSES 8000
#define S_TGT     40
#define SE        81
#define NSLOT     9
#define DEPTH     8
#define LOG_TINY  (-87.336544750402f)   // log(float32 tiny), matches reference

__device__ __forceinline__ float log_add(float a, float b) {
    float x = fmaxf(a, b);
    float y = fminf(a, b);
    return x + log1pf(__expf(y - x));
}

__device__ __forceinline__ float lane_up1(float x, int lane) {
    int src = (lane > 0 ? lane - 1 : lane) << 2;   // byte address of source lane
    return __int_as_float(__builtin_amdgcn_ds_bpermute(src, __float_as_int(x)));
}

// Async DMA: per-lane 4B gather from (saddr + voff) into LDS byte address lds_byte.
__device__ __forceinline__ void async_gather_b32(unsigned lds_byte, unsigned voff,
                                                 unsigned long long sbase) {
    asm volatile("global_load_async_to_lds_b32 %0, %1, %2"
                 :: "v"(lds_byte), "v"(voff), "s"(sbase)
                 : "memory");
}

__global__ __launch_bounds__(32) void ctc_fwd_kernel(
    const float* __restrict__ log_probs,      // (T, N, C)
    const int*   __restrict__ targets,        // (N, S)
    const int*   __restrict__ target_lengths, // (N,)
    float*       __restrict__ out)            // (N,)
{
    __shared__ float G[NSLOT][96];   // gathered lp ring: G[slot][state]
    __shared__ float fin[96];        // final alphas for the reduction

    const int lane = threadIdx.x;    // 32-thread block == one wave
    const int n    = blockIdx.x;
    const int s0   = lane * 3;       // first state owned by this lane

    // ---- per-lane expanded-target class byte-offsets + skip masks -----------
    unsigned coff0 = 0, coff1 = 0, coff2 = 0;
    bool skip0 = false, skip1 = false, skip2 = false;
    {
        const int* tg = targets + n * S_TGT;
        int s, i, cls;
        s = s0;                                     // state s0
        if (s < SE && (s & 1)) { i = (s - 1) >> 1; cls = tg[i];
            coff0 = (unsigned)cls << 2;
            if (s >= 3) skip0 = (cls != tg[i - 1]); }
        s = s0 + 1;                                 // state s0+1
        if (s < SE && (s & 1)) { i = (s - 1) >> 1; cls = tg[i];
            coff1 = (unsigned)cls << 2;
            if (s >= 3) skip1 = (cls != tg[i - 1]); }
        s = s0 + 2;                                 // state s0+2
        if (s < SE && (s & 1)) { i = (s - 1) >> 1; cls = tg[i];
            coff2 = (unsigned)cls << 2;
            if (s >= 3) skip2 = (cls != tg[i - 1]); }
    }

    // ---- alpha init (t = 0) -------------------------------------------------
    const float* lp0 = log_probs + (size_t)n * C_CLASSES;
    float a0 = LOG_TINY, a1 = LOG_TINY, a2 = LOG_TINY;
    if (lane == 0) {
        a0 = lp0[0];            // alpha[0] = lp[0, n, blank]
        a1 = lp0[coff1 >> 2];   // alpha[1] = lp[0, n, targets[n,0]]
    }

    // ---- async-DMA prologue: prefetch t = 1..DEPTH into the LDS ring --------
    const unsigned gbase = (unsigned)(size_t)(void*)&G[0][0];  // LDS byte offset
    const unsigned lmine = gbase + (unsigned)(s0 * 4);
    const unsigned long long stride = (unsigned long long)N_BATCH * C_CLASSES * 4ull;
    unsigned long long sb = (unsigned long long)(size_t)log_probs
                          + (unsigned long long)n * C_CLASSES * 4ull
                          + stride;                             // base of row t=1
#pragma unroll
    for (int t = 1; t <= DEPTH; ++t) {
        unsigned ls = lmine + (unsigned)((t % NSLOT) * 384);
        async_gather_b32(ls,     coff0, sb);
        async_gather_b32(ls + 4, coff1, sb);
        async_gather_b32(ls + 8, coff2, sb);
        sb += stride;
    }
    // sb now points at row t = DEPTH+1 for steady-state prefetch

    // ---- 511-step serial recursion, barrier-free ----------------------------
    for (int t = 1; t < T_STEPS; ++t) {
        // 24 async loads outstanding (t..t+7); retire the 3 oldest => row t ready.
        asm volatile("s_wait_asynccnt 21" ::: "memory");
        const int slot = t % NSLOT;
        float lq0 = G[slot][s0];
        float lq1 = G[slot][s0 + 1];
        float lq2 = G[slot][s0 + 2];
        // Ensure the LDS reads completed before the DMA may re-target slots.
        asm volatile("s_wait_dscnt 0" ::: "memory");

        // Prefetch t+DEPTH (clamped to row T-1 near the end so the outstanding
        // count — and the constant wait immediate — stay exact).
        {
            const int tp = t + DEPTH;
            unsigned ls = lmine + (unsigned)((tp % NSLOT) * 384);
            async_gather_b32(ls,     coff0, sb);
            async_gather_b32(ls + 4, coff1, sb);
            async_gather_b32(ls + 8, coff2, sb);
            if (tp < T_STEPS - 1) sb += stride;
        }

        // alpha[s-1], alpha[s-2] for state s0 come from lane-1 via ds_bpermute.
        float pa2 = lane_up1(a2, lane);   // alpha[3l-1]
        float pa1 = lane_up1(a1, lane);   // alpha[3l-2]

        float n0 = (lane == 0) ? a0 : log_add(a0, pa2);
        if (skip0) n0 = log_add(n0, pa1);
        float n1 = log_add(a1, a0);
        if (skip1) n1 = log_add(n1, pa2);
        float n2 = log_add(a2, a1);
        if (skip2) n2 = log_add(n2, a0);

        a0 = n0 + lq0;
        a1 = n1 + lq1;
        a2 = n2 + lq2;
    }

    // ---- reduction: loss = logadd(alpha[L-1], alpha[L-2]) -------------------
    fin[s0]     = a0;
    fin[s0 + 1] = a1;
    fin[s0 + 2] = a2;
    __syncthreads();
    if (lane == 0) {
        const int tl = target_lengths[n];
        const int L  = 2 * tl + 1;
        float loss = log_add(fin[L - 1], fin[L - 2]);
        out[n] = -loss / (float)tl;
    }
}

extern "C" void kernel_launch(void* const* d_in, const int* in_sizes, int n_in,
                              void* d_out, int out_size, void* d_ws, size_t ws_size,
                              hipStream_t stream) {
    const float* log_probs      = (const float*)d_in[0];
    const int*   targets        = (const int*)d_in[1];
    // d_in[2] = input_lengths: intentionally unused (reference ignores it).
    const int*   target_lengths = (const int*)d_in[3];
    float*       out            = (float*)d_out;

    ctc_fwd_kernel<<<dim3(N_BATCH), dim3(32), 0, stream>>>(
        log_probs, targets, target_lengths, out);

    (void)in_sizes; (void)n_in; (void)out_size; (void)d_ws; (void)ws_size;
}